// TSception_12506944766039
// MI455X (gfx1250) — compile-verified
//
#include <hip/hip_runtime.h>
#include <math.h>

typedef float v2f __attribute__((ext_vector_type(2)));
typedef float v8f __attribute__((ext_vector_type(8)));

#define BN_EPS 1e-5f

__device__ __forceinline__ float gelu_exact(float v) {
    // jax.nn.gelu(approximate=False): 0.5*x*(1+erf(x/sqrt(2)))
    return 0.5f * v * (1.0f + erff(v * 0.7071067811865475f));
}

// ---------------------------------------------------------------------------
// Zero the 64x12 feature accumulator in workspace (ws is poisoned with 0xAA).
// ---------------------------------------------------------------------------
__global__ void tsc_zero_ws(float* f) {
    for (int i = threadIdx.x; i < 64 * 12; i += blockDim.x) f[i] = 0.0f;
}

// ---------------------------------------------------------------------------
// Phase 1: one block per (b,c) row of x. Fused conv -> interp -> BN ->
// pointwise(dm=2, 2 sides) -> exact GELU -> mean over T, partial over C.
// Accumulates into f[64][12] via global float atomics (pre-scaled by 1/(C*T)).
// Feature index: k*4 + side*2 + d  (left d0, left d1, right d0, right d1).
// ---------------------------------------------------------------------------
__global__ __launch_bounds__(256) void tsc_phase1(
    const float* __restrict__ x,
    const float* __restrict__ tw0, const float* __restrict__ tb0,
    const float* __restrict__ tw1, const float* __restrict__ tb1,
    const float* __restrict__ tw2, const float* __restrict__ tb2,
    const float* __restrict__ l_gamma, const float* __restrict__ l_beta,
    const float* __restrict__ l_mean,  const float* __restrict__ l_var,
    const float* __restrict__ l_w,     const float* __restrict__ l_b,
    const float* __restrict__ r_gamma, const float* __restrict__ r_beta,
    const float* __restrict__ r_mean,  const float* __restrict__ r_var,
    const float* __restrict__ r_w,     const float* __restrict__ r_b,
    float* __restrict__ f_accum)
{
    constexpr int T = 4096;
    constexpr int C = 32;

    __shared__ float xs[T];        // 16 KB: one (b,c) row
    __shared__ float wts[112];     // conv taps: 64 + 32 + 16
    __shared__ float blocksum[12];

    const int tid = threadIdx.x;
    const int bc  = blockIdx.x;
    const int b   = bc / C;
    const int c   = bc % C;

    // stage the x row (float4 vectorized: global_load_b128)
    const float4* xin = (const float4*)(x + (size_t)bc * T);
    float4* xs4 = (float4*)xs;
    for (int i = tid; i < T / 4; i += 256) xs4[i] = xin[i];

    if (tid < 64)                    wts[tid] = tw0[tid];
    else if (tid < 96)               wts[tid] = tw1[tid - 64];
    else if (tid < 112)              wts[tid] = tw2[tid - 96];
    if (tid < 12)                    blocksum[tid] = 0.0f;
    __syncthreads();

    // fold BN(eval) + pointwise weight/bias into z = y*A + B per (side,d)
    const float lsc = l_gamma[c] * rsqrtf(l_var[c] + BN_EPS);
    const float lof = l_beta[c] - l_mean[c] * lsc;
    const float rsc = r_gamma[c] * rsqrtf(r_var[c] + BN_EPS);
    const float rof = r_beta[c] - r_mean[c] * rsc;
    float Am[4], Bm[4];
    Am[0] = lsc * l_w[c * 2 + 0]; Bm[0] = lof * l_w[c * 2 + 0] + l_b[c * 2 + 0];
    Am[1] = lsc * l_w[c * 2 + 1]; Bm[1] = lof * l_w[c * 2 + 1] + l_b[c * 2 + 1];
    Am[2] = rsc * r_w[c * 2 + 0]; Bm[2] = rof * r_w[c * 2 + 0] + r_b[c * 2 + 0];
    Am[3] = rsc * r_w[c * 2 + 1]; Bm[3] = rof * r_w[c * 2 + 1] + r_b[c * 2 + 1];

    const int   KS[3]   = {64, 32, 16};
    const int   LK[3]   = {4033, 4065, 4081};
    const int   WOFF[3] = {0, 64, 96};
    const float cb[3]   = {tb0[0], tb1[0], tb2[0]};

    float acc[12];
    #pragma unroll
    for (int i = 0; i < 12; ++i) acc[i] = 0.0f;

    for (int t = tid; t < T; t += 256) {
        #pragma unroll
        for (int s = 0; s < 3; ++s) {
            const int ks = KS[s];
            const int Lk = LK[s];
            const float* __restrict__ w = wts + WOFF[s];
            // F.interpolate(mode='linear', align_corners=False)
            float src = (t + 0.5f) * ((float)Lk / (float)T) - 0.5f;
            src = fminf(fmaxf(src, 0.0f), (float)(Lk - 1));
            const int   i0 = (int)src;                 // floor (src >= 0)
            const int   i1 = min(i0 + 1, Lk - 1);
            const float wf = src - (float)i0;
            // VALID conv evaluated only at the two needed taps
            float s0 = 0.0f, s1 = 0.0f;
            #pragma unroll 16
            for (int j = 0; j < ks; ++j) {
                const float wj = w[j];
                s0 = fmaf(xs[i0 + j], wj, s0);
                s1 = fmaf(xs[i1 + j], wj, s1);
            }
            const float y = (cb[s] + s0) * (1.0f - wf) + (cb[s] + s1) * wf;
            #pragma unroll
            for (int d = 0; d < 4; ++d)
                acc[s * 4 + d] += gelu_exact(fmaf(y, Am[d], Bm[d]));
        }
    }

    // wave32 reduction, then block reduction, then global atomic accumulate
    const int lane = tid & 31;
    #pragma unroll
    for (int i = 0; i < 12; ++i) {
        float v = acc[i];
        #pragma unroll
        for (int off = 16; off > 0; off >>= 1) v += __shfl_xor(v, off, 32);
        if (lane == 0) atomicAdd(&blocksum[i], v);
    }
    __syncthreads();
    if (tid < 12)
        atomicAdd(&f_accum[b * 12 + tid],
                  blocksum[tid] * (1.0f / ((float)C * (float)T)));
}

// ---------------------------------------------------------------------------
// Phase 2: single block, 8 waves. fp32 WMMA 16x16x4 GEMMs:
//   h1 = gelu(f[64,12] @ fw1[12,128] + fb1)
//   h2 =       h1      @ fw2[128,128] + fb2
//   out = (h2 @ vw + vb, h2 @ aw + ab)   -> d_out[0:128], d_out[128:256]
// Fragment layouts per CDNA5 ISA 7.12.2:
//   A 16x4 : lanes 0-15 -> K{0,1}, lanes 16-31 -> K{2,3}; M = lane&15
//   B 4x16 : VGPR v holds row K=v (lanes 0-15) / K=v+2 (lanes 16-31); N = lane&15
//   C/D    : VGPR r holds M=r (lanes 0-15) / M=r+8 (lanes 16-31);     N = lane&15
// ---------------------------------------------------------------------------
__global__ __launch_bounds__(256) void tsc_phase2(
    const float* __restrict__ f,
    const float* __restrict__ fw1, const float* __restrict__ fb1,
    const float* __restrict__ fw2, const float* __restrict__ fb2,
    const float* __restrict__ vw,  const float* __restrict__ vb,
    const float* __restrict__ aw,  const float* __restrict__ ab,
    float* __restrict__ out)
{
    __shared__ float sf[64 * 12];
    __shared__ float h[64 * 128];   // holds h1, then reused for h2

    const int tid  = threadIdx.x;
    const int lane = tid & 31;
    const int wave = tid >> 5;

    for (int i = tid; i < 64 * 12; i += 256) sf[i] = f[i];
    __syncthreads();

    const int mtile = wave >> 1;            // 0..3  (16-row tile)
    const int nbase = (wave & 1) * 4;       // 4 N-tiles per wave
    const int l15   = lane & 15;
    const int khalf = (lane < 16) ? 0 : 2;  // K sub-offset for A/B fragments
    const int mhalf = (lane < 16) ? 0 : 8;  // M offset for C/D rows
    const int arow  = mtile * 16 + l15;

    // ---- layer 1: K = 12 (3 WMMA k-steps) ----
    #pragma unroll
    for (int nt = 0; nt < 4; ++nt) {
        const int ntile = nbase + nt;
        const int col   = ntile * 16 + l15;
        v8f accv = {};
        #pragma unroll
        for (int kst = 0; kst < 3; ++kst) {
            const int kb = kst * 4 + khalf;
            v2f a, bf;
            a.x  = sf[arow * 12 + kb];
            a.y  = sf[arow * 12 + kb + 1];
            bf.x = fw1[kb * 128 + col];
            bf.y = fw1[(kb + 1) * 128 + col];
            accv = __builtin_amdgcn_wmma_f32_16x16x4_f32(
                false, a, false, bf, (short)0, accv, false, false);
        }
        const float bias = fb1[col];
        #pragma unroll
        for (int r = 0; r < 8; ++r)
            h[(mtile * 16 + mhalf + r) * 128 + col] = gelu_exact(accv[r] + bias);
    }
    __syncthreads();

    // ---- layer 2: K = 128 (32 WMMA k-steps), result kept in VGPRs ----
    float h2reg[4][8];
    #pragma unroll
    for (int nt = 0; nt < 4; ++nt) {
        const int ntile = nbase + nt;
        const int col   = ntile * 16 + l15;
        v8f accv = {};
        for (int kst = 0; kst < 32; ++kst) {
            const int kb = kst * 4 + khalf;
            v2f a, bf;
            a.x  = h[arow * 128 + kb];
            a.y  = h[arow * 128 + kb + 1];
            bf.x = fw2[kb * 128 + col];
            bf.y = fw2[(kb + 1) * 128 + col];
            accv = __builtin_amdgcn_wmma_f32_16x16x4_f32(
                false, a, false, bf, (short)0, accv, false, false);
        }
        const float bias = fb2[col];
        #pragma unroll
        for (int r = 0; r < 8; ++r) h2reg[nt][r] = accv[r] + bias;
    }
    __syncthreads();   // all reads of h1 complete before overwrite
    #pragma unroll
    for (int nt = 0; nt < 4; ++nt) {
        const int col = (nbase + nt) * 16 + l15;
        #pragma unroll
        for (int r = 0; r < 8; ++r)
            h[(mtile * 16 + mhalf + r) * 128 + col] = h2reg[nt][r];
    }
    __syncthreads();

    // ---- heads: 64 rows x {v0,v1,a0,a1} -> exactly 256 threads ----
    {
        const int row  = tid >> 2;
        const int j    = tid & 3;
        const int head = j >> 1;
        const int colo = j & 1;
        const float* __restrict__ W = head ? aw : vw;
        float s2 = head ? ab[colo] : vb[colo];
        const float* __restrict__ hr = h + row * 128;
        #pragma unroll 8
        for (int k = 0; k < 128; ++k) s2 = fmaf(hr[k], W[k * 2 + colo], s2);
        out[head * 128 + row * 2 + colo] = s2;
    }
}

// ---------------------------------------------------------------------------
extern "C" void kernel_launch(void* const* d_in, const int* in_sizes, int n_in,
                              void* d_out, int out_size, void* d_ws, size_t ws_size,
                              hipStream_t stream) {
    const float* x       = (const float*)d_in[0];
    const float* tw0     = (const float*)d_in[1];
    const float* tb0     = (const float*)d_in[2];
    const float* tw1     = (const float*)d_in[3];
    const float* tb1     = (const float*)d_in[4];
    const float* tw2     = (const float*)d_in[5];
    const float* tb2     = (const float*)d_in[6];
    const float* l_gamma = (const float*)d_in[7];
    const float* l_beta  = (const float*)d_in[8];
    const float* l_mean  = (const float*)d_in[9];
    const float* l_var   = (const float*)d_in[10];
    const float* l_w     = (const float*)d_in[11];
    const float* l_b     = (const float*)d_in[12];
    const float* r_gamma = (const float*)d_in[13];
    const float* r_beta  = (const float*)d_in[14];
    const float* r_mean  = (const float*)d_in[15];
    const float* r_var   = (const float*)d_in[16];
    const float* r_w     = (const float*)d_in[17];
    const float* r_b     = (const float*)d_in[18];
    const float* fw1     = (const float*)d_in[19];
    const float* fb1     = (const float*)d_in[20];
    const float* fw2     = (const float*)d_in[21];
    const float* fb2     = (const float*)d_in[22];
    const float* vw      = (const float*)d_in[23];
    const float* vb      = (const float*)d_in[24];
    const float* aw      = (const float*)d_in[25];
    const float* ab      = (const float*)d_in[26];

    float* fbuf = (float*)d_ws;       // 64*12 fp32 feature accumulator
    float* outp = (float*)d_out;      // 256 fp32: [64,2] v then [64,2] a

    hipLaunchKernelGGL(tsc_zero_ws, dim3(1), dim3(256), 0, stream, fbuf);
    hipLaunchKernelGGL(tsc_phase1, dim3(64 * 32), dim3(256), 0, stream,
                       x, tw0, tb0, tw1, tb1, tw2, tb2,
                       l_gamma, l_beta, l_mean, l_var, l_w, l_b,
                       r_gamma, r_beta, r_mean, r_var, r_w, r_b, fbuf);
    hipLaunchKernelGGL(tsc_phase2, dim3(1), dim3(256), 0, stream,
                       fbuf, fw1, fb1, fw2, fb2, vw, vb, aw, ab, outp);
}